// ResidualRGCN_57011395887295
// MI455X (gfx1250) — compile-verified
//
#include <hip/hip_runtime.h>
#include <hip/hip_bf16.h>
#include <stddef.h>

#define HD   128
#define RREL 8
#define BBAS 8
#define LLAY 3
#define FDIM 6
#define BN_EPS 1e-5f

typedef float v2f  __attribute__((ext_vector_type(2)));
typedef float v8f  __attribute__((ext_vector_type(8)));
typedef unsigned int u32x4 __attribute__((ext_vector_type(4)));
typedef int   i32x4 __attribute__((ext_vector_type(4)));
typedef int   i32x8 __attribute__((ext_vector_type(8)));

// ---------------------------------------------------------------------------
// TDM: async-load a 16 x 128 f32 tile (row stride HD elements in memory) into
// LDS with a 132-float destination pitch via the TDM pad feature.
//   pad_interval = 6  -> pad after every 128 DWORDs (one tile row)
//   pad_amount   = 3  -> insert 4 DWORDs (row pitch becomes 132 floats)
// D# packing per CDNA5 ISA.
// ---------------------------------------------------------------------------
__device__ __forceinline__ void tdm_load_tile16x128(
    void* lds_dst, const float* gsrc, int rows_remaining)
{
    unsigned lds_off = (unsigned)(size_t)lds_dst;         // LDS aperture: low 32 bits
    unsigned long long ga = (unsigned long long)(size_t)gsrc;

    u32x4 g0;
    g0[0] = 1u;                                           // count=1 (valid user D#)
    g0[1] = lds_off;                                      // lds_addr
    g0[2] = (unsigned)(ga & 0xFFFFFFFFu);                 // global_addr[31:0]
    g0[3] = (unsigned)((ga >> 32) & 0x1FFFFFFu)           // global_addr[56:32]
          | (2u << 30);                                   // type = 2 ("image")

    i32x8 g1;
    g1[0] = (2 << 16)      // data_size = 4B
          | (1 << 20)      // pad_enable
          | (6 << 22)      // pad_interval: every 128 DWORDs
          | (3 << 25);     // pad_amount: 4 DWORDs
    g1[1] = (int)((HD & 0xFFFFu) << 16);                  // tensor_dim0[15:0] = 128
    g1[2] = (int)(((unsigned)rows_remaining & 0xFFFFu) << 16); // dim0 hi=0 | tensor_dim1 lo
    g1[3] = (int)(HD << 16);                              // dim1 hi=0 | tile_dim0 = 128
    g1[4] = 16;                                           // tile_dim1 = 16, tile_dim2 = 0
    g1[5] = HD;                                           // tensor_dim0_stride[31:0] = 128
    g1[6] = 0;                                            // stride hi | dim1_stride lo
    g1[7] = 0;

    i32x4 gz = {0, 0, 0, 0};
#if defined(__clang_major__) && (__clang_major__ >= 23)
    i32x8 gz8 = {0, 0, 0, 0, 0, 0, 0, 0};
    __builtin_amdgcn_tensor_load_to_lds(g0, g1, gz, gz, gz8, 0);
#else
    __builtin_amdgcn_tensor_load_to_lds(g0, g1, gz, gz, 0);
#endif
}

// ---------------------------------------------------------------------------
// x = emb[x_ids] + relu(node_feat @ feat_w^T + feat_b) + virtual
// one thread per (node, 4 channels): b128 traffic
// ---------------------------------------------------------------------------
__global__ __launch_bounds__(256) void k_init_x(
    const int* __restrict__ x_ids, const float* __restrict__ node_feat,
    const float* __restrict__ emb, const float* __restrict__ feat_w,
    const float* __restrict__ feat_b, const float* __restrict__ virt,
    float* __restrict__ x, int N)
{
    int idx = blockIdx.x * 256 + threadIdx.x;   // N * HD/4 threads
    if (idx >= N * (HD / 4)) return;
    int n  = idx >> 5;
    int c0 = (idx & 31) * 4;

    float nf[FDIM];
#pragma unroll
    for (int f = 0; f < FDIM; ++f) nf[f] = node_feat[n * FDIM + f];

    const float4 e4 = *(const float4*)(emb + (size_t)x_ids[n] * HD + c0);
    const float4 v4 = *(const float4*)(virt + c0);
    float4 o;
#pragma unroll
    for (int j = 0; j < 4; ++j) {
        int c = c0 + j;
        float acc = feat_b[c];
#pragma unroll
        for (int f = 0; f < FDIM; ++f)
            acc = fmaf(nf[f], feat_w[c * FDIM + f], acc);
        ((float*)&o)[j] = ((const float*)&e4)[j] + fmaxf(acc, 0.0f)
                        + ((const float*)&v4)[j];
    }
    *(float4*)(x + (size_t)n * HD + c0) = o;
}

// ---------------------------------------------------------------------------
// per-(node,relation) edge counts, then invert
// ---------------------------------------------------------------------------
__global__ __launch_bounds__(256) void k_count_edges(
    const int* __restrict__ dst, const int* __restrict__ etype,
    float* __restrict__ cnt, int E)
{
    int e = blockIdx.x * 256 + threadIdx.x;
    if (e >= E) return;
    atomicAdd(&cnt[dst[e] * RREL + etype[e]], 1.0f);
}

__global__ __launch_bounds__(256) void k_inv_counts(float* __restrict__ cnt, int NR)
{
    int i = blockIdx.x * 256 + threadIdx.x;
    if (i >= NR) return;
    cnt[i] = 1.0f / fmaxf(cnt[i], 1.0f);
}

// ---------------------------------------------------------------------------
// WT[r][o][i] = sum_b comp[r,b] * bases[b][i][o]   (transposed, r = 0..7)
// WT[8][o][i] = root[i][o]                          (transposed root)
// Transposed storage makes each lane's WMMA B-fragment pair contiguous
// (one global_load_b64 per WMMA instead of two b32 loads).
// ---------------------------------------------------------------------------
__global__ __launch_bounds__(256) void k_make_WT(
    const float* __restrict__ comp, const float* __restrict__ bases,
    const float* __restrict__ root, float* __restrict__ WT)
{
    int idx = blockIdx.x * 256 + threadIdx.x;   // (R+1)*H*H = 147456 total
    if (idx >= (RREL + 1) * HD * HD) return;
    int r   = idx >> 14;            // /(H*H)
    int rem = idx & (HD * HD - 1);
    int o   = rem >> 7;
    int i   = rem & (HD - 1);
    float acc;
    if (r < RREL) {
        acc = 0.0f;
#pragma unroll
        for (int b = 0; b < BBAS; ++b)
            acc = fmaf(comp[r * BBAS + b], bases[b * HD * HD + i * HD + o], acc);
    } else {
        acc = root[i * HD + o];
    }
    WT[idx] = acc;
}

// ---------------------------------------------------------------------------
// scatter: sums[r, dst, :] += x[src, :]
// one thread per (edge, 4 channels): b128 gather + 4 atomics, indices
// amortized over 4 channels.
// ---------------------------------------------------------------------------
__global__ __launch_bounds__(256) void k_scatter(
    const int* __restrict__ src, const int* __restrict__ dst,
    const int* __restrict__ etype, const float* __restrict__ x,
    float* __restrict__ sums, int N, int E)
{
    long long idx = (long long)blockIdx.x * 256 + threadIdx.x;
    if (idx >= (long long)E * (HD / 4)) return;
    int e  = (int)(idx >> 5);           // HD/4 = 32 quads per edge
    int c0 = (int)(idx & 31) * 4;
    int s = src[e], d = dst[e], r = etype[e];
    const float4 v = *(const float4*)(x + (size_t)s * HD + c0);
    float* base = sums + ((size_t)r * N + d) * HD + c0;
    atomicAdd(base + 0, v.x);
    atomicAdd(base + 1, v.y);
    atomicAdd(base + 2, v.z);
    atomicAdd(base + 3, v.w);
}

// ---------------------------------------------------------------------------
// Fused GEMM:  h = sum_r (inv_cnt_r * sums_r) @ W[r]  +  x @ root  +  bias
//
// Block = 256 threads = 8 waves; covers a 16-node row tile x full H=128.
// Each wave32 owns one 16x16 tile: 32 x V_WMMA_F32_16X16X4_F32 per B matrix,
// 9 B matrices (8 relations + root), accumulator resident in VGPRs.
// A tiles are DMA'd global->LDS by the Tensor Data Mover, double-buffered:
// wave 0 issues tensor_load_to_lds for relation r+1 while all waves compute
// relation r, then drains TENSORcnt to 1 (s_wait_tensorcnt).
// inv_cnt row scaling commutes with the GEMM -> applied on A-fragment read.
// ---------------------------------------------------------------------------
__global__ __launch_bounds__(256) void k_layer_gemm(
    const float* __restrict__ sums,     // [R,N,H]
    const float* __restrict__ x,        // [N,H]
    const float* __restrict__ WT,       // [R+1,H,H] transposed weights
    const float* __restrict__ bias,     // [H]
    const float* __restrict__ inv_cnt,  // [N*R]
    float* __restrict__ hout,           // [N,H]
    int N)
{
    __shared__ float At[2][16][132];    // double-buffered A tile, padded pitch

    const int tid  = threadIdx.x;
    const int wave = tid >> 5;          // 0..7 -> 16-col slice
    const int lane = tid & 31;
    const int lm   = lane & 15;         // A row / B column within tile
    const int hi   = lane >> 4;         // 0: K pair {0,1}; 1: K pair {2,3}
    const int row0 = blockIdx.x * 16;
    const int col0 = wave * 16;

    v8f acc = {};

    // prologue: kick off DMA of relation 0's A tile
    if (wave == 0)
        tdm_load_tile16x128(&At[0][0][0],
                            sums + (size_t)row0 * HD, N - row0);

#pragma unroll 1
    for (int r = 0; r < RREL + 1; ++r) {
        if (wave == 0) {
            if (r < RREL) {
                // prefetch next A tile (relation r+1, or the x tile for root)
                const float* nsrc = (r + 1 < RREL)
                    ? sums + ((size_t)(r + 1) * N + row0) * HD
                    : x + (size_t)row0 * HD;
                tdm_load_tile16x128(&At[(r + 1) & 1][0][0], nsrc, N - row0);
                __builtin_amdgcn_s_wait_tensorcnt(1);  // tile r landed
            } else {
                __builtin_amdgcn_s_wait_tensorcnt(0);  // last tile landed
            }
        }
        __syncthreads();                // LDS tile r visible to all waves

        const float s = (r < RREL) ? inv_cnt[(row0 + lm) * RREL + r] : 1.0f;
        const float* Bl = WT + ((size_t)r * HD + (col0 + lm)) * HD;
        const float (*A)[132] = At[r & 1];

#pragma unroll
        for (int k = 0; k < HD; k += 4) {
            int ka = k + hi * 2;
            v2f a = *(const v2f*)&A[lm][ka];
            a.x *= s;
            a.y *= s;
            v2f b = *(const v2f*)(Bl + ka);
            acc = __builtin_amdgcn_wmma_f32_16x16x4_f32(
                false, a, false, b, (short)0, acc, false, false);
        }
        __syncthreads();                // all reads of tile r done before reuse
    }

    // store + bias: VGPR i -> rows (row0+i) and (row0+i+8)
    const int c = col0 + lm;
#pragma unroll
    for (int i = 0; i < 8; ++i) {
        int n = row0 + i + hi * 8;
        if (n < N) hout[(size_t)n * HD + c] = acc[i] + bias[c];
    }
}

// ---------------------------------------------------------------------------
// BatchNorm statistics (sum, sumsq per channel over N nodes)
// ---------------------------------------------------------------------------
__global__ __launch_bounds__(256) void k_bn_stats(
    const float* __restrict__ h, float* __restrict__ s, float* __restrict__ ss,
    int N)
{
    int c    = threadIdx.x & (HD - 1);
    int part = threadIdx.x >> 7;        // 0..1
    float sum = 0.0f, sumsq = 0.0f;
    for (int n = blockIdx.x * 2 + part; n < N; n += gridDim.x * 2) {
        float v = h[(size_t)n * HD + c];
        sum += v;
        sumsq = fmaf(v, v, sumsq);
    }
    atomicAdd(&s[c], sum);
    atomicAdd(&ss[c], sumsq);
}

// ---------------------------------------------------------------------------
// x += relu(gamma * (h - mu) * rsqrt(var + eps) + beta), float4-vectorized
// ---------------------------------------------------------------------------
__global__ __launch_bounds__(256) void k_bn_apply_residual(
    const float* __restrict__ h, const float* __restrict__ s,
    const float* __restrict__ ss, const float* __restrict__ gamma,
    const float* __restrict__ beta, float* __restrict__ x, int N)
{
    int idx = blockIdx.x * 256 + threadIdx.x;   // N * HD/4 threads
    if (idx >= N * (HD / 4)) return;
    int c0 = (idx & 31) * 4;
    size_t off = (size_t)idx * 4;

    const float4 h4 = *(const float4*)(h + off);
    float4 x4 = *(const float4*)(x + off);
    float invN = 1.0f / (float)N;
#pragma unroll
    for (int j = 0; j < 4; ++j) {
        int c = c0 + j;
        float mu  = s[c] * invN;
        float var = ss[c] * invN - mu * mu;
        float v = gamma[c] * (((const float*)&h4)[j] - mu) * rsqrtf(var + BN_EPS)
                + beta[c];
        ((float*)&x4)[j] += fmaxf(v, 0.0f);
    }
    *(float4*)(x + off) = x4;
}

// ---------------------------------------------------------------------------
extern "C" void kernel_launch(void* const* d_in, const int* in_sizes, int n_in,
                              void* d_out, int out_size, void* d_ws, size_t ws_size,
                              hipStream_t stream)
{
    const int*   x_ids     = (const int*)d_in[0];
    const int*   edge_idx  = (const int*)d_in[1];   // [2,E]
    const int*   edge_type = (const int*)d_in[2];
    const float* node_feat = (const float*)d_in[3];
    const float* emb       = (const float*)d_in[4];
    const float* feat_w    = (const float*)d_in[5];
    const float* feat_b    = (const float*)d_in[6];
    const float* virt      = (const float*)d_in[7];
    const float* bases     = (const float*)d_in[8];   // [L,B,H,H]
    const float* comp      = (const float*)d_in[9];   // [L,R,B]
    const float* root      = (const float*)d_in[10];  // [L,H,H]
    const float* bias      = (const float*)d_in[11];  // [L,H]
    const float* gamma     = (const float*)d_in[12];  // [L,H]
    const float* beta      = (const float*)d_in[13];  // [L,H]

    const int N = in_sizes[0];
    const int E = in_sizes[2];
    const int* e_src = edge_idx;
    const int* e_dst = edge_idx + E;

    float* x = (float*)d_out;                    // x lives in d_out [N,H]

    // workspace layout (floats)
    float* ws    = (float*)d_ws;
    float* sums  = ws;                                       // R*N*H
    float* hbuf  = sums + (size_t)RREL * N * HD;             // N*H
    float* WTbuf = hbuf + (size_t)N * HD;                    // (R+1)*H*H
    float* cnt   = WTbuf + (size_t)(RREL + 1) * HD * HD;     // N*R (cnt -> inv_cnt)
    float* bns   = cnt + (size_t)N * RREL;                   // H
    float* bnss  = bns + HD;                                 // H

    const int NH4_blocks = (N * (HD / 4) + 255) / 256;
    const int row_tiles  = (N + 15) / 16;

    // init x
    k_init_x<<<NH4_blocks, 256, 0, stream>>>(x_ids, node_feat, emb, feat_w,
                                             feat_b, virt, x, N);

    // edge counts -> inv_cnt
    hipMemsetAsync(cnt, 0, (size_t)N * RREL * sizeof(float), stream);
    k_count_edges<<<(E + 255) / 256, 256, 0, stream>>>(e_dst, edge_type, cnt, E);
    k_inv_counts<<<(N * RREL + 255) / 256, 256, 0, stream>>>(cnt, N * RREL);

    for (int l = 0; l < LLAY; ++l) {
        // relation weights (basis decomposition) + root, both transposed
        k_make_WT<<<((RREL + 1) * HD * HD + 255) / 256, 256, 0, stream>>>(
            comp + (size_t)l * RREL * BBAS, bases + (size_t)l * BBAS * HD * HD,
            root + (size_t)l * HD * HD, WTbuf);

        // scatter raw x into per-(relation,node) sums
        hipMemsetAsync(sums, 0, (size_t)RREL * N * HD * sizeof(float), stream);
        long long tot = (long long)E * (HD / 4);
        k_scatter<<<(unsigned)((tot + 255) / 256), 256, 0, stream>>>(
            e_src, e_dst, edge_type, x, sums, N, E);

        // fused 9-GEMM (8 relations + root) with TDM-fed WMMA, + bias -> hbuf
        k_layer_gemm<<<row_tiles, 256, 0, stream>>>(
            sums, x, WTbuf, bias + (size_t)l * HD, cnt, hbuf, N);

        // batch norm + relu + residual
        hipMemsetAsync(bns, 0, 2 * HD * sizeof(float), stream);
        k_bn_stats<<<512, 256, 0, stream>>>(hbuf, bns, bnss, N);
        k_bn_apply_residual<<<NH4_blocks, 256, 0, stream>>>(
            hbuf, bns, bnss, gamma + (size_t)l * HD, beta + (size_t)l * HD, x, N);
    }
}